// ReEig_13108240187439
// MI455X (gfx1250) — compile-verified
//
#include <hip/hip_runtime.h>

// ---------------------------------------------------------------------------
// ReEig via matrix-sign Newton-Schulz, all-WMMA (V_WMMA_F32_16X16X4_F32).
//   M = X - eps*I
//   Y0 = M / ||M||_F ;  Y <- 1.5*Y - 0.5*Y*(Y*Y)   (NIT times)  => sign(M)
//   Out = eps*I + 0.5*(M + M @ sign(M))
// One workgroup (8 wave32) per 64x64 matrix, matrix resident in LDS.
// ---------------------------------------------------------------------------

typedef __attribute__((ext_vector_type(2))) float v2f;
typedef __attribute__((ext_vector_type(8))) float v8f;

#define MATN    64
#define LSTR    68          // LDS row stride in floats: 4-bank skew, 16B aligned
#define NIT     18
#define REEPS   1e-4f

// C = A(64x64) @ B(64x64) for one 16x16 tile, fp32 WMMA, K accumulated 4 at a time.
__device__ __forceinline__ v8f gemm_tile(const float* __restrict__ Abuf,
                                         const float* __restrict__ Bbuf,
                                         int tm, int tn, int lane)
{
    const int half = lane >> 4;          // 0: lanes 0-15, 1: lanes 16-31
    const int l    = lane & 15;
    const int arow = tm * 16 + l;        // A-fragment: M = lane%16
    const int bcol = tn * 16 + l;        // B-fragment: N = lane%16
    v8f acc = {};
#pragma unroll
    for (int kt = 0; kt < 16; ++kt) {
        const int k0 = kt * 4;
        // A 16x4 fragment: v0 = A[M][k0 + 2*half], v1 = A[M][k0 + 2*half + 1]
        const float* ap = Abuf + arow * LSTR + k0 + half * 2;
        v2f a;  a.x = ap[0];  a.y = ap[1];
        // B 4x16 fragment: v0 = B[k0 + 2*half][N], v1 = B[k0 + 2*half + 1][N]
        const float* bp = Bbuf + (k0 + half * 2) * LSTR + bcol;
        v2f b;  b.x = bp[0];  b.y = bp[LSTR];
        acc = __builtin_amdgcn_wmma_f32_16x16x4_f32(
                  false, a, false, b, (short)0, acc, false, false);
    }
    return acc;
}

__global__ __launch_bounds__(256)
void reeig_sign_kernel(const float* __restrict__ X,
                       float* __restrict__ Out)
{
    __shared__ float lM[MATN * LSTR];    // M = X - eps*I   (unscaled)
    __shared__ float lY[MATN * LSTR];    // Newton-Schulz iterate -> sign(M)
    __shared__ float lT[MATN * LSTR];    // Y*Y scratch
    __shared__ float red;

    const int tid  = threadIdx.x;
    const int lane = tid & 31;
    const int wave = tid >> 5;

    const float* Xb = X   + (size_t)blockIdx.x * (MATN * MATN);
    float*       Ob = Out + (size_t)blockIdx.x * (MATN * MATN);

    if (tid == 0) red = 0.0f;
    __syncthreads();

    // ---- load X, form M = X - eps*I, accumulate Frobenius^2 ----------------
    float ss = 0.0f;
#pragma unroll
    for (int i = 0; i < 4; ++i) {
        const int q = tid + i * 256;             // float4 index, 16 per row
        const int r = q >> 4;
        const int c = (q & 15) << 2;
        float4 v = ((const float4*)Xb)[q];
        if (r >= c && r < c + 4) ((float*)&v)[r - c] -= REEPS;  // diagonal hit
        ss += v.x * v.x + v.y * v.y + v.z * v.z + v.w * v.w;
        *(float4*)&lM[r * LSTR + c] = v;
    }
    atomicAdd(&red, ss);                          // ds_add_f32
    __syncthreads();

    const float fro = sqrtf(red);
    const float inv = (fro > 0.0f) ? (1.0f / fro) : 0.0f;

    // ---- Y0 = M / ||M||_F --------------------------------------------------
#pragma unroll
    for (int i = 0; i < 4; ++i) {
        const int q = tid + i * 256;
        const int r = q >> 4;
        const int c = (q & 15) << 2;
        float4 v = *(const float4*)&lM[r * LSTR + c];
        v.x *= inv; v.y *= inv; v.z *= inv; v.w *= inv;
        *(float4*)&lY[r * LSTR + c] = v;
    }
    __syncthreads();

    // Each wave owns 2 of the 16 output tiles.
    const int t0  = wave;                 // tiles 0..7
    const int t1  = wave + 8;             // tiles 8..15
    const int tm0 = t0 >> 2, tn0 = t0 & 3;
    const int tm1 = t1 >> 2, tn1 = t1 & 3;
    const int half = lane >> 4;
    const int l    = lane & 15;

    // ---- Newton-Schulz: Y <- 1.5*Y - 0.5*Y*(Y*Y) ---------------------------
    for (int it = 0; it < NIT; ++it) {
        // T = Y @ Y
        v8f c0 = gemm_tile(lY, lY, tm0, tn0, lane);
        v8f c1 = gemm_tile(lY, lY, tm1, tn1, lane);
#pragma unroll
        for (int v = 0; v < 8; ++v) {
            lT[(tm0 * 16 + half * 8 + v) * LSTR + (tn0 * 16 + l)] = c0[v];
            lT[(tm1 * 16 + half * 8 + v) * LSTR + (tn1 * 16 + l)] = c1[v];
        }
        __syncthreads();

        // G = Y @ T  (into registers), then in-place Y <- 1.5*Y - 0.5*G
        v8f g0 = gemm_tile(lY, lT, tm0, tn0, lane);
        v8f g1 = gemm_tile(lY, lT, tm1, tn1, lane);
        __syncthreads();                  // all reads of Y complete
#pragma unroll
        for (int v = 0; v < 8; ++v) {
            const int i0 = (tm0 * 16 + half * 8 + v) * LSTR + (tn0 * 16 + l);
            const int i1 = (tm1 * 16 + half * 8 + v) * LSTR + (tn1 * 16 + l);
            lY[i0] = 1.5f * lY[i0] - 0.5f * g0[v];
            lY[i1] = 1.5f * lY[i1] - 0.5f * g1[v];
        }
        __syncthreads();
    }

    // ---- Out = eps*I + 0.5*(M + M @ sign(M)) -------------------------------
    v8f r0 = gemm_tile(lM, lY, tm0, tn0, lane);
    v8f r1 = gemm_tile(lM, lY, tm1, tn1, lane);
#pragma unroll
    for (int v = 0; v < 8; ++v) {
        {
            const int row = tm0 * 16 + half * 8 + v;
            const int col = tn0 * 16 + l;
            float o = 0.5f * (lM[row * LSTR + col] + r0[v]);
            if (row == col) o += REEPS;
            Ob[row * MATN + col] = o;
        }
        {
            const int row = tm1 * 16 + half * 8 + v;
            const int col = tn1 * 16 + l;
            float o = 0.5f * (lM[row * LSTR + col] + r1[v]);
            if (row == col) o += REEPS;
            Ob[row * MATN + col] = o;
        }
    }
}

extern "C" void kernel_launch(void* const* d_in, const int* in_sizes, int n_in,
                              void* d_out, int out_size, void* d_ws, size_t ws_size,
                              hipStream_t stream) {
    const float* X = (const float*)d_in[0];
    float* Out = (float*)d_out;
    const int B = in_sizes[0] / (MATN * MATN);   // 8192
    reeig_sign_kernel<<<dim3(B), dim3(256), 0, stream>>>(X, Out);
}